// BATPredictor_50362786512900
// MI455X (gfx1250) — compile-verified
//
#include <hip/hip_runtime.h>
#include <math.h>
#include <stdint.h>

#define B_ 16
#define S_ 2048
#define C_ 512

// K1 LDS staging geometry
#define CHUNK 32            // channels per staged chunk
#define ROWS 130            // 128 s-rows + 1-row halo each side
#define RPAD 136            // padded row stride (2*136 % 64 == 16 -> conflict-free halves)
#define NELEM (CHUNK * ROWS)
#define ITERS ((NELEM + 255) / 256)   // 17 async loads per thread per chunk

typedef float v2f __attribute__((ext_vector_type(2)));
typedef float v8f __attribute__((ext_vector_type(8)));

// gfx1250 async global->LDS copy (ASYNCcnt-tracked). Inline asm: portable
// across ROCm-7.2 / amdgpu-toolchain builtin arity differences.
__device__ __forceinline__ void async_ld_b32(uint32_t lds_off, const void* g)
{
    asm volatile("global_load_async_to_lds_b32 %0, %1, off"
                 :: "v"(lds_off), "v"((unsigned long long)(uintptr_t)g)
                 : "memory");
}
__device__ __forceinline__ void wait_async_le(int) {}  // (doc helper, unused)
__device__ __forceinline__ void wait_async_17()
{
    asm volatile("s_wait_asynccnt 0x11" ::: "memory");  // allow the 17 just-issued
}
__device__ __forceinline__ void wait_async_0()
{
    asm volatile("s_wait_asynccnt 0x0" ::: "memory");
}

// ---------------------------------------------------------------------------
// K1: async-LDS-staged depthwise conv(K=3) + residual + ReLU + GEMV via
//     V_WMMA_F32_16X16X4_F32 + sigmoid -> alphas.
//     Block: 256 thr = 8 waves; wave w owns s-rows [s_blk+16w, s_blk+16w+16).
//     WMMA: A[m][k]=lin_w[cb+k] (broadcast over m), B[k][n]=y[cb+k][s0+n],
//     D row 0 accumulates the 16 dot products exactly in f32.
// ---------------------------------------------------------------------------
__global__ __launch_bounds__(256) void k1_alphas(
    const float* __restrict__ hidden, const float* __restrict__ conv_w,
    const float* __restrict__ conv_b, const float* __restrict__ lin_w,
    const float* __restrict__ lin_b, float* __restrict__ alphas_ws,
    float* __restrict__ out_alpha, float* __restrict__ out_peak)
{
    __shared__ float tile[2][CHUNK * RPAD];   // 2 x 17.4 KB, transposed [c][r]

    const int tid   = threadIdx.x;
    const int lane  = tid & 31;
    const int wave  = tid >> 5;
    const int bid   = blockIdx.x;             // 256 blocks
    const int b     = bid >> 4;
    const int s_blk = (bid & 15) << 7;        // 128 rows per block
    const int n     = lane & 15;
    const int h     = lane >> 4;
    const int s     = s_blk + wave * 16 + n;

    const float mm = (s > 0)      ? 1.0f : 0.0f;   // halo masks (staged rows clamped)
    const float mp = (s < S_ - 1) ? 1.0f : 0.0f;
    const int rbase = wave * 16 + n;               // r_local of row s-1

    const float* gbase = hidden + (size_t)b * S_ * C_;

    // Stage chunk cc -> buffer buf. Exactly ITERS async instrs per wave
    // (tail clamped to a benign duplicate), so waits can use a fixed count.
    auto stage = [&](int cc, int buf) {
        const uint32_t lbase = (uint32_t)(uintptr_t)&tile[buf][0];
        #pragma unroll
        for (int k = 0; k < ITERS; ++k) {
            int idx = k * 256 + tid;
            int c = idx & (CHUNK - 1);
            int r = idx >> 5;                      // idx / CHUNK
            if (r > ROWS - 1) r = ROWS - 1;        // benign duplicate on tail
            int rg = s_blk - 1 + r;
            rg = rg < 0 ? 0 : (rg > S_ - 1 ? S_ - 1 : rg);
            async_ld_b32(lbase + (uint32_t)(c * RPAD + r) * 4u,
                         gbase + (size_t)rg * C_ + cc + c);
        }
    };

    v8f acc = {0.f, 0.f, 0.f, 0.f, 0.f, 0.f, 0.f, 0.f};

    stage(0, 0);
    for (int chunk = 0; chunk < C_ / CHUNK; ++chunk) {
        const int cc = chunk * CHUNK;
        if (chunk < C_ / CHUNK - 1) {
            stage(cc + CHUNK, (chunk + 1) & 1);    // prefetch next chunk
            wait_async_17();                       // current chunk complete
        } else {
            wait_async_0();
        }
        __syncthreads();

        const float* tb = tile[chunk & 1];
        #pragma unroll 4
        for (int cl = 0; cl < CHUNK; cl += 4) {
            const int c0l = cl + 2 * h;            // lane's channel pair (local)
            const int c0  = cc + c0l;
            v2f a;
            a.x = lin_w[c0];
            a.y = lin_w[c0 + 1];
            const float w0a = conv_w[c0 * 3 + 0], w1a = conv_w[c0 * 3 + 1], w2a = conv_w[c0 * 3 + 2];
            const float w0b = conv_w[c0 * 3 + 3], w1b = conv_w[c0 * 3 + 4], w2b = conv_w[c0 * 3 + 5];

            // transposed tile: taps (s-1,s,s+1) are contiguous rows
            const float* t0 = tb + c0l * RPAD + rbase;
            const float* t1 = t0 + RPAD;
            const float hm0 = t0[0], hc0 = t0[1], hp0 = t0[2];
            const float hm1 = t1[0], hc1 = t1[1], hp1 = t1[2];

            float y0 = fmaf(w0a, hm0 * mm, fmaf(w1a, hc0, fmaf(w2a, hp0 * mp, conv_b[c0]))) + hc0;
            float y1 = fmaf(w0b, hm1 * mm, fmaf(w1b, hc1, fmaf(w2b, hp1 * mp, conv_b[c0 + 1]))) + hc1;
            v2f bv;
            bv.x = fmaxf(y0, 0.0f);
            bv.y = fmaxf(y1, 0.0f);

            acc = __builtin_amdgcn_wmma_f32_16x16x4_f32(
                false, a, false, bv, (short)0, acc, false, false);
        }
        __syncthreads();                           // before next stage overwrites
    }

    const float logit = acc[0] + lin_b[0];
    const float alpha = 1.0f / (1.0f + expf(-logit));

    if (h == 0) {
        const int idx = b * S_ + s;
        alphas_ws[idx] = alpha;
        out_alpha[idx] = alpha;
        out_peak[idx]  = alpha;
    }
}

// ---------------------------------------------------------------------------
// K2: per-batch inclusive cumsum of alphas (wave32 shfl scan) + token_num.
// ---------------------------------------------------------------------------
__global__ __launch_bounds__(32) void k2_scan(
    const float* __restrict__ alphas, float* __restrict__ csum,
    float* __restrict__ out_tok)
{
    const int b = blockIdx.x;
    const int lane = threadIdx.x;
    float carry = 0.0f;
    for (int chunk = 0; chunk < S_ / 32; ++chunk) {
        const int idx = b * S_ + chunk * 32 + lane;
        float v = alphas[idx];
#pragma unroll
        for (int d = 1; d < 32; d <<= 1) {
            float t = __shfl_up(v, d, 32);
            if (lane >= d) v += t;
        }
        v += carry;
        csum[idx] = v;
        carry = __shfl(v, 31, 32);
    }
    if (lane == 0) out_tok[b] = carry;
}

// ---------------------------------------------------------------------------
// K3: row-start table  A[b][t] = first s with csum >= t  (sentinel S_ = none).
// ---------------------------------------------------------------------------
__global__ void k3_init(int* __restrict__ Arr)
{
    const int i = blockIdx.x * blockDim.x + threadIdx.x;
    if (i < B_ * (S_ + 1)) Arr[i] = S_;
}

__global__ void k3_bounds(const float* __restrict__ csum, int* __restrict__ Arr)
{
    const int i = blockIdx.x * blockDim.x + threadIdx.x;
    if (i >= B_ * S_) return;
    const int b = i / S_;
    const int s = i - b * S_;
    const float c  = csum[i];
    const float cp = (s > 0) ? csum[i - 1] : 0.0f;
    const int r  = (int)floorf(c);
    const int rp = (int)floorf(cp);
    if (s == 0) Arr[b * (S_ + 1)] = 0;            // a_0 = 0 always
    if (r > rp && r < S_) Arr[b * (S_ + 1) + r] = s;
}

// ---------------------------------------------------------------------------
// K4: deterministic gather (atomic-free CIF). One block per (b,t).
// ---------------------------------------------------------------------------
__global__ __launch_bounds__(256) void k4_gather(
    const float* __restrict__ hidden, const float* __restrict__ alphas,
    const float* __restrict__ csum, const int* __restrict__ Arr,
    float* __restrict__ out)
{
    const int t   = blockIdx.x;
    const int b   = blockIdx.y;
    const int tid = threadIdx.x;
    const int a  = Arr[b * (S_ + 1) + t];
    const int a1 = Arr[b * (S_ + 1) + t + 1];

    float2 acc = make_float2(0.0f, 0.0f);
    if (a < S_) {
        const int send = min(a1, S_ - 1);
        const float csA = csum[b * S_ + a];
        for (int s = a; s <= send; ++s) {
            float w;
            if (s == a)        w = csA - (float)t;
            else if (s == a1)  w = (float)(t + 1) - csum[b * S_ + s - 1];
            else               w = alphas[b * S_ + s];
            const float2* hp = (const float2*)(hidden + ((size_t)(b * S_ + s)) * C_);
            if (s < send)
                __builtin_prefetch(hidden + ((size_t)(b * S_ + s + 1)) * C_ + tid * 2, 0, 0);
            const float2 v = hp[tid];
            acc.x = fmaf(w, v.x, acc.x);
            acc.y = fmaf(w, v.y, acc.y);
        }
    }
    ((float2*)(out + ((size_t)(b * S_ + t)) * C_))[tid] = acc;
}

// ---------------------------------------------------------------------------
extern "C" void kernel_launch(void* const* d_in, const int* in_sizes, int n_in,
                              void* d_out, int out_size, void* d_ws, size_t ws_size,
                              hipStream_t stream)
{
    const float* hidden = (const float*)d_in[0];
    const float* conv_w = (const float*)d_in[1];
    const float* conv_b = (const float*)d_in[2];
    const float* lin_w  = (const float*)d_in[3];
    const float* lin_b  = (const float*)d_in[4];
    float* out = (float*)d_out;

    // output layout: acoustic (B,S,C) | token_num (B) | alphas (B,S) | cif_peak (B,S)
    const size_t off_tok   = (size_t)B_ * S_ * C_;
    const size_t off_alpha = off_tok + B_;
    const size_t off_peak  = off_alpha + (size_t)B_ * S_;

    float* alphas_ws = (float*)d_ws;                   // B*S floats
    float* csum      = alphas_ws + (size_t)B_ * S_;    // B*S floats
    int*   Arr       = (int*)(csum + (size_t)B_ * S_); // B*(S+1) ints

    k1_alphas<<<256, 256, 0, stream>>>(hidden, conv_w, conv_b, lin_w, lin_b,
                                       alphas_ws, out + off_alpha, out + off_peak);
    k2_scan<<<B_, 32, 0, stream>>>(alphas_ws, csum, out + off_tok);
    k3_init<<<(B_ * (S_ + 1) + 255) / 256, 256, 0, stream>>>(Arr);
    k3_bounds<<<(B_ * S_ + 255) / 256, 256, 0, stream>>>(csum, Arr);
    dim3 g4(S_, B_);
    k4_gather<<<g4, 256, 0, stream>>>(hidden, alphas_ws, csum, Arr, out);
}